// TreeSelfAttentionGPU_30116310680241
// MI455X (gfx1250) — compile-verified
//
#include <hip/hip_runtime.h>
#include <math.h>

typedef __attribute__((ext_vector_type(2))) float v2f;
typedef __attribute__((ext_vector_type(8))) float v8f;

#define DD 1024
#define TWOD 2048
#define BATCH 4
#define TSEQ 8192
#define NNODES 1023

// ---------------------------------------------------------------------------
// Kernel 1: leaf means. node_states[b][511+j][d] = mean(x[b, 8j:8j+8, d]), j<512
// ---------------------------------------------------------------------------
__global__ void __launch_bounds__(256)
leaves_kernel(const float* __restrict__ x, float* __restrict__ ns) {
    const int leaf = blockIdx.x;   // 0..511
    const int b    = blockIdx.y;   // 0..3
    const int tid  = threadIdx.x;  // 0..255
    const float* xp = x  + ((size_t)b * TSEQ + (size_t)leaf * 8) * DD;
    float*       np = ns + ((size_t)b * NNODES + (511 + leaf)) * DD;
    #pragma unroll
    for (int k = 0; k < 4; ++k) {
        const int d = tid + k * 256;
        float s = 0.f;
        #pragma unroll
        for (int t = 0; t < 8; ++t) s += xp[(size_t)t * DD + d];
        np[d] = s * 0.125f;
    }
}

// ---------------------------------------------------------------------------
// Kernel 2 (per tree level): proj = [L,R] @ W^T via V_WMMA_F32_16X16X4_F32,
// then thresholded spike gate: out = proj > bt[level] ? proj : 0.
// Each wave computes TWO 16x16 M-tiles (32 rows) sharing every B load, giving
// two independent WMMA accumulator chains (better ILP, fewer loads per WMMA).
// The 4 waves of a block share A addresses (same M rows) -> WGP$ hits.
// Rows r = b*NL + j map to node (level_start + j) of batch b.
// ---------------------------------------------------------------------------
__global__ void __launch_bounds__(128)
tree_level_kernel(const float* __restrict__ W,   // (1024, 2048) row-major [d][e]
                  const float* __restrict__ bt,  // (10, 1024)
                  float* __restrict__ ns,        // (4, 1023, 1024)
                  int level, int NL, int level_start) {
    const int lane = threadIdx.x;          // 0..31
    const int wv   = threadIdx.y;          // 0..3
    const int M    = BATCH * NL;

    const int nBase = (blockIdx.x * 4 + wv) * 16;   // N-tile base (0..1008)
    const int col   = lane & 15;                    // N within tile
    const int kb    = (lane < 16) ? 0 : 2;          // K sub-offset per ISA layout
    const int mA    = lane & 15;                    // A-matrix row for this lane

    // Per-lane A rows for the two M-tiles (clamped; stores are guarded below).
    int rA0 = blockIdx.y * 32 + mA;
    int rA1 = rA0 + 16;
    if (rA0 > M - 1) rA0 = M - 1;
    if (rA1 > M - 1) rA1 = M - 1;

    const int bA0 = rA0 / NL, jA0 = rA0 - bA0 * NL, n0 = level_start + jA0;
    const int bA1 = rA1 / NL, jA1 = rA1 - bA1 * NL, n1 = level_start + jA1;
    const float* l0 = ns + ((size_t)bA0 * NNODES + (2 * n0 + 1)) * DD + kb;
    const float* r0 = ns + ((size_t)bA0 * NNODES + (2 * n0 + 2)) * DD + kb;
    const float* l1 = ns + ((size_t)bA1 * NNODES + (2 * n1 + 1)) * DD + kb;
    const float* r1 = ns + ((size_t)bA1 * NNODES + (2 * n1 + 2)) * DD + kb;
    const float* wp = W  + (size_t)(nBase + col) * TWOD + kb;

    v8f c0 = {};
    v8f c1 = {};
    // First half of K: left children (e = 0..1023)
    #pragma unroll 4
    for (int e = 0; e < DD; e += 4) {
        v2f bb = *(const v2f*)(wp + e);
        v2f a0 = *(const v2f*)(l0 + e);
        v2f a1 = *(const v2f*)(l1 + e);
        c0 = __builtin_amdgcn_wmma_f32_16x16x4_f32(false, a0, false, bb,
                                                   (short)0, c0, false, false);
        c1 = __builtin_amdgcn_wmma_f32_16x16x4_f32(false, a1, false, bb,
                                                   (short)0, c1, false, false);
    }
    // Second half of K: right children (e = 1024..2047)
    #pragma unroll 4
    for (int e = 0; e < DD; e += 4) {
        v2f bb = *(const v2f*)(wp + DD + e);
        v2f a0 = *(const v2f*)(r0 + e);
        v2f a1 = *(const v2f*)(r1 + e);
        c0 = __builtin_amdgcn_wmma_f32_16x16x4_f32(false, a0, false, bb,
                                                   (short)0, c0, false, false);
        c1 = __builtin_amdgcn_wmma_f32_16x16x4_f32(false, a1, false, bb,
                                                   (short)0, c1, false, false);
    }

    // Threshold gate + store (C/D layout: M = v (+8 upper half), N = lane&15)
    const float btv = bt[(size_t)level * DD + nBase + col];
    #pragma unroll
    for (int v = 0; v < 8; ++v) {
        const int mrow = (lane < 16) ? v : v + 8;
        // tile 0
        {
            const int r = blockIdx.y * 32 + mrow;
            if (r < M) {
                const int b = r / NL;
                const int j = r - b * NL;
                const int n = level_start + j;
                float val = c0[v];
                val = (val > btv) ? val : 0.f;
                ns[((size_t)b * NNODES + n) * DD + nBase + col] = val;
            }
        }
        // tile 1
        {
            const int r = blockIdx.y * 32 + 16 + mrow;
            if (r < M) {
                const int b = r / NL;
                const int j = r - b * NL;
                const int n = level_start + j;
                float val = c1[v];
                val = (val > btv) ? val : 0.f;
                ns[((size_t)b * NNODES + n) * DD + nBase + col] = val;
            }
        }
    }
}

// ---------------------------------------------------------------------------
// Kernel 3: column softmax over node_weights (axis = node) fused with
// mixture[b][d] = sum_n softmax(nw)[n][d] * ns[b][n][d]  (online softmax).
// Threads map to consecutive d -> fully coalesced reads.
// ---------------------------------------------------------------------------
__global__ void __launch_bounds__(256)
mixture_kernel(const float* __restrict__ nw, const float* __restrict__ ns,
               float* __restrict__ mix) {
    const int d = blockIdx.x * 256 + threadIdx.x;  // 0..1023
    float m = -INFINITY, s = 0.f;
    float a0 = 0.f, a1 = 0.f, a2 = 0.f, a3 = 0.f;
    for (int n = 0; n < NNODES; ++n) {
        const float w  = nw[(size_t)n * DD + d];
        const float nm = fmaxf(m, w);
        const float sc = __expf(m - nm);   // first iter: exp(-inf) = 0
        const float e  = __expf(w - nm);
        s  = s  * sc + e;
        a0 = a0 * sc + e * ns[((size_t)0 * NNODES + n) * DD + d];
        a1 = a1 * sc + e * ns[((size_t)1 * NNODES + n) * DD + d];
        a2 = a2 * sc + e * ns[((size_t)2 * NNODES + n) * DD + d];
        a3 = a3 * sc + e * ns[((size_t)3 * NNODES + n) * DD + d];
        m = nm;
    }
    const float inv = 1.f / s;
    mix[0 * DD + d] = a0 * inv;
    mix[1 * DD + d] = a1 * inv;
    mix[2 * DD + d] = a2 * inv;
    mix[3 * DD + d] = a3 * inv;
}

// ---------------------------------------------------------------------------
// Kernel 4: out[b][t][:] = rmsnorm(mixture[b] + x[b][t]) * rms_w
// One block per (b,t) row; float4 vectorized; bandwidth-bound pass.
// ---------------------------------------------------------------------------
__global__ void __launch_bounds__(256)
finalize_kernel(const float* __restrict__ x, const float* __restrict__ mix,
                const float* __restrict__ rms_w, float* __restrict__ out) {
    const int row = blockIdx.x;           // 0..32767
    const int b   = row >> 13;            // T = 8192
    const int t   = row & (TSEQ - 1);
    const int tid = threadIdx.x;          // 0..255, 4 floats each

    const float4* xp = (const float4*)(x + ((size_t)b * TSEQ + t) * DD);
    const float4* mp = (const float4*)(mix + (size_t)b * DD);
    const float4 xv = xp[tid];
    const float4 mv = mp[tid];
    float4 o;
    o.x = xv.x + mv.x; o.y = xv.y + mv.y; o.z = xv.z + mv.z; o.w = xv.w + mv.w;
    float ss = o.x * o.x + o.y * o.y + o.z * o.z + o.w * o.w;

    __shared__ float red[256];
    __shared__ float rinv_s;
    red[tid] = ss;
    __syncthreads();
    for (int off = 128; off > 0; off >>= 1) {
        if (tid < off) red[tid] += red[tid + off];
        __syncthreads();
    }
    if (tid == 0) rinv_s = rsqrtf(red[0] * (1.0f / DD) + 1.1920929e-07f);
    __syncthreads();
    const float r = rinv_s;

    const float4 g = ((const float4*)rms_w)[tid];
    float4 ov;
    ov.x = o.x * r * g.x; ov.y = o.y * r * g.y;
    ov.z = o.z * r * g.z; ov.w = o.w * r * g.w;
    ((float4*)(out + ((size_t)b * TSEQ + t) * DD))[tid] = ov;
}

// ---------------------------------------------------------------------------
extern "C" void kernel_launch(void* const* d_in, const int* in_sizes, int n_in,
                              void* d_out, int out_size, void* d_ws, size_t ws_size,
                              hipStream_t stream) {
    const float* x     = (const float*)d_in[0];  // (4, 8192, 1024)
    const float* W     = (const float*)d_in[1];  // (1024, 2048)
    const float* nw    = (const float*)d_in[2];  // (1023, 1024)
    const float* bt    = (const float*)d_in[3];  // (10, 1024)
    // d_in[4] = as_w: multiplied by 0.0 in the reference -> dead input
    const float* rms_w = (const float*)d_in[5];  // (1024,)
    float* out = (float*)d_out;

    float* ns  = (float*)d_ws;                          // 4*1023*1024 floats
    float* mix = ns + (size_t)BATCH * NNODES * DD;      // 4*1024 floats

    // 1) Leaf means into node slots 511..1022 (nodes 0..510 are overwritten
    //    by the tree passes, so no zero-init of the workspace is needed).
    leaves_kernel<<<dim3(512, BATCH), 256, 0, stream>>>(x, ns);

    // 2) Tree combine, levels 8..0 (level 9 is masked out in the reference).
    for (int level = 8; level >= 0; --level) {
        const int NL = 1 << level;
        const int level_start = NL - 1;
        const int M = BATCH * NL;
        dim3 grid(16, (M + 31) / 32);      // 16 blk * 4 waves = 64 N-tiles
        tree_level_kernel<<<grid, dim3(32, 4), 0, stream>>>(
            W, bt, ns, level, NL, level_start);
    }

    // 3) softmax over nodes + mixture
    mixture_kernel<<<DD / 256, 256, 0, stream>>>(nw, ns, mix);

    // 4) broadcast-add + RMS norm (bandwidth-bound final pass)
    finalize_kernel<<<BATCH * TSEQ, 256, 0, stream>>>(x, mix, rms_w, out);
}